// WordClassifier_20263655703051
// MI455X (gfx1250) — compile-verified
//
#include <hip/hip_runtime.h>
#include <math.h>

// ---------------------------------------------------------------------------
// Problem constants (from reference): B=16, L=512, H=768, S1=32, S2=128, NS=8, M=4
#define Bc   16
#define Lc   512
#define Hc   768
#define S1c  32
#define S2c  128
#define NSc  8
#define Mc   4
#define H4c  (Hc / 4)   // 192 float4 per row

typedef __attribute__((ext_vector_type(2))) float v2f;
typedef __attribute__((ext_vector_type(8))) float v8f;

// ---------------------------------------------------------------------------
// Block-wide sum reduction over 256 threads
__device__ __forceinline__ float block_reduce_sum(float v, float* red) {
    const int t = threadIdx.x;
    red[t] = v;
    __syncthreads();
    for (int off = 128; off > 0; off >>= 1) {
        if (t < off) red[t] += red[t + off];
        __syncthreads();
    }
    float r = red[0];
    __syncthreads();
    return r;
}

// ---------------------------------------------------------------------------
// 1) row_cls[r,h] = sum_s mask * (e1+e2)[r,s,h] / sum_s mask     r in [0,512)
//    192 threads, one float4 column slice per thread -> global_load_b128,
//    wave-uniform mask branch skips ~half the HBM traffic.
__global__ void k_row_cls(const float* __restrict__ e1, const float* __restrict__ e2,
                          const int* __restrict__ am, float* __restrict__ row_cls) {
    const int r = blockIdx.x;                     // 0..511
    __shared__ float m[S2c];
    for (int s = threadIdx.x; s < S2c; s += 192) m[s] = (float)am[r * S2c + s];
    __syncthreads();
    float cnt = 0.f;
    #pragma unroll 8
    for (int s = 0; s < S2c; ++s) cnt += m[s];
    const float inv = 1.f / cnt;

    const float4* p1 = reinterpret_cast<const float4*>(e1 + (size_t)r * S2c * Hc) + threadIdx.x;
    const float4* p2 = reinterpret_cast<const float4*>(e2 + (size_t)r * S2c * Hc) + threadIdx.x;
    float4 acc = make_float4(0.f, 0.f, 0.f, 0.f);
    for (int s = 0; s < S2c; ++s) {
        if (m[s] != 0.f) {                        // uniform across the wave
            const float4 a = p1[s * H4c];
            const float4 b = p2[s * H4c];
            acc.x += a.x + b.x; acc.y += a.y + b.y;
            acc.z += a.z + b.z; acc.w += a.w + b.w;
        }
    }
    acc.x *= inv; acc.y *= inv; acc.z *= inv; acc.w *= inv;
    reinterpret_cast<float4*>(row_cls + (size_t)r * Hc)[threadIdx.x] = acc;
}

// ---------------------------------------------------------------------------
// 2) sp[b,h] = masked mean over L of (sent_h1+sent_h2) -- float4 streaming
__global__ void k_sp(const float* __restrict__ h1, const float* __restrict__ h2,
                     const int* __restrict__ am, float* __restrict__ sp) {
    const int b = blockIdx.x;
    __shared__ float m[Lc];
    for (int s = threadIdx.x; s < Lc; s += 192) m[s] = (float)am[b * Lc + s];
    __syncthreads();
    float cnt = 0.f;
    for (int s = 0; s < Lc; ++s) cnt += m[s];
    const float inv = 1.f / cnt;

    const float4* p1 = reinterpret_cast<const float4*>(h1 + (size_t)b * Lc * Hc) + threadIdx.x;
    const float4* p2 = reinterpret_cast<const float4*>(h2 + (size_t)b * Lc * Hc) + threadIdx.x;
    float4 acc = make_float4(0.f, 0.f, 0.f, 0.f);
    for (int s = 0; s < Lc; ++s) {
        if (m[s] != 0.f) {
            const float4 a = p1[s * H4c];
            const float4 c = p2[s * H4c];
            acc.x += a.x + c.x; acc.y += a.y + c.y;
            acc.z += a.z + c.z; acc.w += a.w + c.w;
        }
    }
    acc.x *= inv; acc.y *= inv; acc.z *= inv; acc.w *= inv;
    reinterpret_cast<float4*>(sp + (size_t)b * Hc)[threadIdx.x] = acc;
}

// ---------------------------------------------------------------------------
// 3) WMMA f32 GEMM:  C[M,N] = A[M,K] * B  (+bias)
//    TRANS_B=0: B is [K,N] row-major.  TRANS_B=1: B element (k,n) = Bsrc[n*K+k].
//    One wave (32 threads) computes one 16x16 tile via V_WMMA_F32_16X16X4_F32.
template <bool TRANS_B, bool BIAS>
__global__ void k_wmma_gemm_f32(const float* __restrict__ A, const float* __restrict__ Bsrc,
                                const float* __restrict__ bias, float* __restrict__ C,
                                int M, int N, int K) {
    const int lane = threadIdx.x & 31;
    const int half = lane >> 4;       // 0 or 1
    const int l    = lane & 15;
    const int n0 = blockIdx.x * 16;
    const int m0 = blockIdx.y * 16;

    v8f acc = {};
    const float* arow = A + (size_t)(m0 + l) * K + 2 * half;  // A: lane l -> row m0+l
    #pragma unroll 8
    for (int k = 0; k < K; k += 4) {
        v2f a;
        a[0] = arow[k];          // K = k + 2*half
        a[1] = arow[k + 1];      // K = k + 2*half + 1
        v2f b;
        if (TRANS_B) {
            const float* bp = Bsrc + (size_t)(n0 + l) * K + k + 2 * half;
            b[0] = bp[0];
            b[1] = bp[1];
        } else {
            b[0] = Bsrc[(size_t)(k + 2 * half) * N + n0 + l];
            b[1] = Bsrc[(size_t)(k + 2 * half + 1) * N + n0 + l];
        }
        // (neg_a, A, neg_b, B, c_mod, C, reuse_a, reuse_b)
        acc = __builtin_amdgcn_wmma_f32_16x16x4_f32(false, a, false, b,
                                                    (short)0, acc, false, false);
    }
    const int col = n0 + l;
    const float bv = BIAS ? bias[col] : 0.f;
    #pragma unroll
    for (int v = 0; v < 8; ++v) {
        const int row = m0 + v + 8 * half;
        C[(size_t)row * N + col] = acc[v] + bv;
    }
}

// ---------------------------------------------------------------------------
// 4) per (b,n): sim_m = dot(tmp[b], cls[b,n,m]) / sqrt(H); masked softmax over m;
//    pooled[b,n,:] = sum_m w_m * cls[b,n,m,:]
__global__ void k_attnpool(const float* __restrict__ tmp, const float* __restrict__ row_cls,
                           const int* __restrict__ eidx, float* __restrict__ pooled) {
    const int bn = blockIdx.x;           // 0..127
    const int b = bn >> 3, n = bn & 7;
    __shared__ float red[256];
    __shared__ float w[Mc];
    const float* tb = tmp + b * Hc;
    float dot[Mc];
    #pragma unroll
    for (int m = 0; m < Mc; ++m) {
        const float* c = row_cls + (size_t)(b * S1c + n * Mc + m) * Hc;
        float acc = 0.f;
        for (int h = threadIdx.x; h < Hc; h += 256) acc += tb[h] * c[h];
        dot[m] = block_reduce_sum(acc, red);
    }
    if (threadIdx.x == 0) {
        const int vcnt = eidx[b * NSc + n];
        const float scale = 0.03608439182435161f;   // 1/sqrt(768)
        float s[Mc], mx = -1e30f;
        #pragma unroll
        for (int m = 0; m < Mc; ++m) {
            s[m] = (m < vcnt) ? dot[m] * scale : -1e30f;
            mx = fmaxf(mx, s[m]);
        }
        float sum = 0.f;
        #pragma unroll
        for (int m = 0; m < Mc; ++m) { s[m] = expf(s[m] - mx); sum += s[m]; }
        #pragma unroll
        for (int m = 0; m < Mc; ++m) w[m] = s[m] / sum;
    }
    __syncthreads();
    for (int h = threadIdx.x; h < Hc; h += 256) {
        float acc = 0.f;
        #pragma unroll
        for (int m = 0; m < Mc; ++m)
            acc += w[m] * row_cls[(size_t)(b * S1c + n * Mc + m) * Hc + h];
        pooled[(size_t)bn * Hc + h] = acc;
    }
}

// ---------------------------------------------------------------------------
// 5) Row LayerNorm over H=768 (3 elems/thread, 256 threads). Safe in-place.
__global__ void k_layernorm(const float* __restrict__ X, const float* __restrict__ gamma,
                            const float* __restrict__ beta, float* __restrict__ Y) {
    const int r = blockIdx.x;
    __shared__ float red[256];
    const float* x = X + (size_t)r * Hc;
    const int t = threadIdx.x;
    float v0 = x[t], v1 = x[t + 256], v2 = x[t + 512];
    const float mean = block_reduce_sum(v0 + v1 + v2, red) * (1.f / Hc);
    const float d0 = v0 - mean, d1 = v1 - mean, d2 = v2 - mean;
    const float var = block_reduce_sum(d0 * d0 + d1 * d1 + d2 * d2, red) * (1.f / Hc);
    const float inv = rsqrtf(var + 1e-12f);
    float* y = Y + (size_t)r * Hc;
    y[t]       = d0 * inv * gamma[t]       + beta[t];
    y[t + 256] = d1 * inv * gamma[t + 256] + beta[t + 256];
    y[t + 512] = d2 * inv * gamma[t + 512] + beta[t + 512];
}

// ---------------------------------------------------------------------------
// 6) span mean: sel[b,h] = mean over s in [start,end) of hidden[b,s,h]  (float4)
__global__ void k_span_mean(const float* __restrict__ hs, const int* __restrict__ st,
                            const int* __restrict__ en, float* __restrict__ sel) {
    const int b = blockIdx.x;
    const int s0 = st[b], s1 = en[b];
    const float inv = 1.f / (float)(s1 - s0);
    const float4* p = reinterpret_cast<const float4*>(hs + (size_t)b * Lc * Hc) + threadIdx.x;
    float4 acc = make_float4(0.f, 0.f, 0.f, 0.f);
    for (int s = s0; s < s1; ++s) {
        const float4 a = p[s * H4c];
        acc.x += a.x; acc.y += a.y; acc.z += a.z; acc.w += a.w;
    }
    acc.x *= inv; acc.y *= inv; acc.z *= inv; acc.w *= inv;
    reinterpret_cast<float4*>(sel + (size_t)b * Hc)[threadIdx.x] = acc;
}

// ---------------------------------------------------------------------------
// 7) cosine score across the NS axis: out[b,h]
__global__ void k_score(const float* __restrict__ sel, const float* __restrict__ cls_out,
                        float* __restrict__ out) {
    const int i = blockIdx.x * blockDim.x + threadIdx.x;
    if (i >= Bc * Hc) return;
    const int b = i / Hc, h = i - b * Hc;
    const float s = sel[i];
    float sumc = 0.f, ssq = 0.f;
    #pragma unroll
    for (int n = 0; n < NSc; ++n) {
        const float c = cls_out[(size_t)(b * NSc + n) * Hc + h];
        sumc += c;
        ssq  += c * c;
    }
    const float n1 = 2.8284271247461903f * fabsf(s);   // sqrt(NS)*|s|
    const float n2 = sqrtf(ssq);
    out[i] = (s * sumc) / (fmaxf(n1, 1e-8f) * fmaxf(n2, 1e-8f));
}

// ---------------------------------------------------------------------------
extern "C" void kernel_launch(void* const* d_in, const int* in_sizes, int n_in,
                              void* d_out, int out_size, void* d_ws, size_t ws_size,
                              hipStream_t stream) {
    (void)in_sizes; (void)n_in; (void)out_size; (void)ws_size;

    const float* hidden  = (const float*)d_in[0];   // [16,512,768]
    const float* sent_h1 = (const float*)d_in[1];
    const float* sent_h2 = (const float*)d_in[2];
    const int*   attn    = (const int*)d_in[3];     // [16,512]
    const float* e1      = (const float*)d_in[4];   // [512,128,768]
    const float* e2      = (const float*)d_in[5];
    const int*   eam     = (const int*)d_in[6];     // [16,32,128]
    const int*   eidx    = (const int*)d_in[7];     // [16,8]
    const int*   estart  = (const int*)d_in[8];     // [16]
    const int*   eend    = (const int*)d_in[9];     // [16]
    const float* W_attn  = (const float*)d_in[10];  // [768,768]
    const float* W_t     = (const float*)d_in[11];  // [768,768]
    const float* b_t     = (const float*)d_in[12];  // [768]
    const float* gamma_t = (const float*)d_in[13];
    const float* beta_t  = (const float*)d_in[14];
    const float* gamma_s = (const float*)d_in[15];
    const float* beta_s  = (const float*)d_in[16];

    float* ws = (float*)d_ws;
    float* row_cls = ws;                                   // 512*768
    float* sp      = row_cls + (size_t)Bc * S1c * Hc;      // 16*768
    float* tmp     = sp + Bc * Hc;                         // 16*768
    float* pooled  = tmp + Bc * Hc;                        // 128*768
    float* Z       = pooled + (size_t)Bc * NSc * Hc;       // 128*768
    float* cls_out = Z + (size_t)Bc * NSc * Hc;            // 128*768
    float* sel     = cls_out + (size_t)Bc * NSc * Hc;      // 16*768

    // Explanation row pooling (dominant streamed traffic; b128 loads, mask-skip)
    k_row_cls<<<Bc * S1c, 192, 0, stream>>>(e1, e2, eam, row_cls);
    // Sentence pooling
    k_sp<<<Bc, 192, 0, stream>>>(sent_h1, sent_h2, attn, sp);
    // tmp = sp @ W_attn   (16x768x768, WMMA f32)
    k_wmma_gemm_f32<false, false><<<dim3(Hc / 16, Bc / 16), 32, 0, stream>>>(
        sp, W_attn, nullptr, tmp, Bc, Hc, Hc);
    // ragged softmax attention pooling
    k_attnpool<<<Bc * NSc, 256, 0, stream>>>(tmp, row_cls, eidx, pooled);
    // Z = pooled @ W_t^T + b_t   (128x768x768, WMMA f32)
    k_wmma_gemm_f32<true, true><<<dim3(Hc / 16, (Bc * NSc) / 16), 32, 0, stream>>>(
        pooled, W_t, b_t, Z, Bc * NSc, Hc, Hc);
    // cls_output = LayerNorm(Z)
    k_layernorm<<<Bc * NSc, 256, 0, stream>>>(Z, gamma_t, beta_t, cls_out);
    // sel = LayerNorm(span mean of hidden_states)
    k_span_mean<<<Bc, 192, 0, stream>>>(hidden, estart, eend, sel);
    k_layernorm<<<Bc, 256, 0, stream>>>(sel, gamma_s, beta_s, sel);
    // cosine score over NS axis
    k_score<<<(Bc * Hc + 255) / 256, 256, 0, stream>>>(sel, cls_out, (float*)d_out);
}